// DNN_SKalmanNet_SLAM_38027640439429
// MI455X (gfx1250) — compile-verified
//
#include <hip/hip_runtime.h>

typedef float v2f __attribute__((ext_vector_type(2)));
typedef float v8f __attribute__((ext_vector_type(8)));

#define NTHREADS 512
#define NWAVES   (NTHREADS / 32)

__device__ __forceinline__ float sigmoidf_(float v) { return 1.f / (1.f + expf(-v)); }

__device__ __forceinline__ v2f ld2(const float* p) {
  v2f r; r.x = p[0]; r.y = p[1]; return r;
}

__device__ __forceinline__ v8f wmma4(v2f a, v2f b, v8f c) {
  return __builtin_amdgcn_wmma_f32_16x16x4_f32(false, a, false, b, (short)0, c,
                                               false, false);
}

// y[m] = act(W[m,:] . x + bias[m]); W row-major [M x K]; xl in LDS zero-padded
// to ceil4(K). Waves own 16-row tiles strided by NWAVES. B column 0 carries x:
// lane 0 reads xl[k..k+1], lane 16 reads xl[k+2..k+3], all other lanes read a
// zero pad -> branch-free, EXEC stays full at every WMMA.
__device__ void gemv_wmma(const float* __restrict__ W,
                          const float* __restrict__ bias,
                          const float* xl, const float* zpad, float* y,
                          int M, int K, bool relu, int wave, int lane) {
  const int Kfull   = K & ~3;           // full 4-chunks
  const int Kgrp    = Kfull & ~15;      // unroll-by-4 groups
  const int ntiles  = (M + 15) >> 4;
  const bool hi     = (lane >= 16);
  const int kbase   = hi ? 2 : 0;
  const bool active = (lane == 0) || (lane == 16);
  const float* xsel0 = (lane == 0) ? xl : (lane == 16 ? (xl + 2) : zpad);
  const int xs4  = active ? 4  : 0;
  const int xs16 = active ? 16 : 0;

  for (int t = wave; t < ntiles; t += NWAVES) {
    const int row0 = t << 4;
    int r = row0 + (lane & 15);
    if (r > M - 1) r = M - 1;                       // clamp, keep EXEC full
    const float* wp = W + (size_t)r * (size_t)K + kbase;
    {
      int rn = row0 + (NWAVES << 4) + (lane & 15);  // prefetch next tile row
      if (rn > M - 1) rn = M - 1;
      __builtin_prefetch(W + (size_t)rn * (size_t)K, 0, 1);
    }
    const float* xsel = xsel0;
    v8f c0 = {0.f, 0.f, 0.f, 0.f, 0.f, 0.f, 0.f, 0.f};
    v8f c1 = {0.f, 0.f, 0.f, 0.f, 0.f, 0.f, 0.f, 0.f};
    int k0 = 0;
    for (; k0 < Kgrp; k0 += 16) {                   // 4 chunks per iter
      c0 = wmma4(ld2(wp + 0),  ld2(xsel + 0),  c0);
      c1 = wmma4(ld2(wp + 4),  ld2(xsel + 4),  c1);
      c0 = wmma4(ld2(wp + 8),  ld2(xsel + 8),  c0);
      c1 = wmma4(ld2(wp + 12), ld2(xsel + 12), c1);
      wp += 16; xsel += xs16;
    }
    for (; k0 < Kfull; k0 += 4) {                   // leftover full chunks
      c0 = wmma4(ld2(wp), ld2(xsel), c0);
      wp += 4; xsel += xs4;
    }
    if (Kfull < K) {                                // guarded K tail
      v2f a;
      const int ka = Kfull + kbase;
      a.x = (ka     < K) ? wp[0] : 0.f;
      a.y = (ka + 1 < K) ? wp[1] : 0.f;
      c0 = wmma4(a, ld2(xsel), c0);                 // x pads are zero
    }
#pragma unroll
    for (int i = 0; i < 8; ++i) c0[i] += c1[i];
    if (active) {                                   // lanes 0/16 hold column 0
      const int base = row0 + (hi ? 8 : 0);
#pragma unroll
      for (int i = 0; i < 8; ++i) {
        const int m = base + i;
        if (m < M) {
          float v = c0[i] + bias[m];
          y[m] = relu ? fmaxf(v, 0.f) : v;
        }
      }
    }
  }
}

struct BranchP {
  const float *Wl1, *bl1;
  const float *wih0, *whh0, *bih0, *bhh0;
  const float *wih1, *whh1, *bih1, *bhh1;
  const float *Wl2a, *bl2a, *Wl2b, *bl2b;
  const float *hn;      // [2 x 87]
  const float *ia, *ib; // innovation / diff inputs
  int na, nb, K1, Mout, out_off;
};
struct Params {
  BranchP br[2];
  const float *lin, *jac;
  float* out;
};

__global__ __launch_bounds__(NTHREADS)
void skalman_fused(Params P) {
  const BranchP bp = P.br[blockIdx.x];
  const int tid  = threadIdx.x;
  const int wave = tid >> 5;
  const int lane = tid & 31;

  __shared__ float inv[24];           // concat input, zero-padded to 24
  __shared__ float xv[500];           // l1 output (K=500, already 4-aligned)
  __shared__ float gi[261], gh[261];  // GRU gate pre-activations
  __shared__ float hA[88], hB[88], hC[88];
  __shared__ float pv[40];
  __shared__ float zpad[16];          // zero source for inactive B lanes

  // ---- stage inputs into LDS ----
  if (tid < 16) zpad[tid] = 0.f;
  if (tid < bp.na) inv[tid] = bp.ia[tid];
  if (tid < bp.nb) inv[bp.na + tid] = bp.ib[tid];
  if (tid < 2)  inv[bp.na + bp.nb + tid] = P.lin[tid];
  if (tid < 10) inv[bp.na + bp.nb + 2 + tid] = P.jac[tid];
  if (tid >= bp.K1 && tid < 24) inv[tid] = 0.f;
  if (tid < 87) { hA[tid] = bp.hn[tid]; hB[tid] = bp.hn[87 + tid]; }
  if (tid == 87) { hA[87] = 0.f; hB[87] = 0.f; hC[87] = 0.f; }
  __syncthreads();

  // ---- x = relu(W_l1 @ in + b_l1) : [500] ----
  gemv_wmma(bp.Wl1, bp.bl1, inv, zpad, xv, 500, bp.K1, true, wave, lane);
  __syncthreads();

  // ---- GRU cell 0: gi = wih0@x + bih0 ; gh = whh0@h0 + bhh0 ----
  gemv_wmma(bp.wih0, bp.bih0, xv, zpad, gi, 261, 500, false, wave, lane);
  gemv_wmma(bp.whh0, bp.bhh0, hA, zpad, gh, 261,  87, false, wave, lane);
  __syncthreads();
  if (tid < 87) {
    float r = sigmoidf_(gi[tid]      + gh[tid]);
    float z = sigmoidf_(gi[87 + tid] + gh[87 + tid]);
    float n = tanhf(gi[174 + tid] + r * gh[174 + tid]);
    hC[tid] = (1.f - z) * n + z * hA[tid];
  }
  __syncthreads();

  // ---- GRU cell 1 ----
  gemv_wmma(bp.wih1, bp.bih1, hC, zpad, gi, 261, 87, false, wave, lane);
  gemv_wmma(bp.whh1, bp.bhh1, hB, zpad, gh, 261, 87, false, wave, lane);
  __syncthreads();
  if (tid < 87) {
    float r = sigmoidf_(gi[tid]      + gh[tid]);
    float z = sigmoidf_(gi[87 + tid] + gh[87 + tid]);
    float n = tanhf(gi[174 + tid] + r * gh[174 + tid]);
    hA[tid] = (1.f - z) * n + z * hB[tid];        // hA[87] pad still 0
  }
  __syncthreads();

  // ---- p = relu(W_l2a @ h + b_l2a) : [40] ----
  gemv_wmma(bp.Wl2a, bp.bl2a, hA, zpad, pv, 40, 87, true, wave, lane);
  __syncthreads();

  // ---- out = W_l2b @ p + b_l2b -> global ----
  gemv_wmma(bp.Wl2b, bp.bl2b, pv, zpad, P.out + bp.out_off, bp.Mout, 40, false,
            wave, lane);
}

extern "C" void kernel_launch(void* const* d_in, const int* in_sizes, int n_in,
                              void* d_out, int out_size, void* d_ws,
                              size_t ws_size, hipStream_t stream) {
  (void)in_sizes; (void)n_in; (void)out_size; (void)d_ws; (void)ws_size;
  auto f = [&](int i) { return (const float*)d_in[i]; };
  Params P;
  // Branch 1 (Pk): inputs 0..19, hn1 = 34
  P.br[0] = { f(6),  f(7),  f(8),  f(9),  f(10), f(11), f(12), f(13), f(14),
              f(15), f(16), f(17), f(18), f(19), f(34), f(0),  f(2),
              5, 5, 22, 25, 0 };
  // Branch 2 (Sk): inputs 20..33, hn2 = 35
  P.br[1] = { f(20), f(21), f(22), f(23), f(24), f(25), f(26), f(27), f(28),
              f(29), f(30), f(31), f(32), f(33), f(35), f(1),  f(3),
              2, 2, 16, 4, 25 };
  P.lin = f(4);
  P.jac = f(5);
  P.out = (float*)d_out;
  skalman_fused<<<dim3(2), dim3(NTHREADS), 0, stream>>>(P);
}